// MyGNN_88845693485538
// MI455X (gfx1250) — compile-verified
//
#include <hip/hip_runtime.h>
#include <hip/hip_bf16.h>

typedef __attribute__((ext_vector_type(2))) float v2f;
typedef __attribute__((ext_vector_type(8))) float v8f;

#define DIN  128
#define DHID 64
#define DOUT 2

// ---------------- degree / normalization ----------------

__global__ void k_deg_init(float* __restrict__ deg, int N) {
  int i = blockIdx.x * blockDim.x + threadIdx.x;
  if (i < N) deg[i] = 1.0f;  // self-loop weight
}

__global__ void k_deg_edges(const long long* __restrict__ ei,
                            const float* __restrict__ ew,
                            float* __restrict__ deg, int E) {
  int e = blockIdx.x * blockDim.x + threadIdx.x;
  if (e < E) {
    int dst = (int)ei[(long long)E + e];
    atomicAdd(&deg[dst], ew[e]);
  }
}

__global__ void k_dinv(float* __restrict__ dinv, int N) {
  int i = blockIdx.x * blockDim.x + threadIdx.x;
  if (i < N) {
    float d = dinv[i];
    dinv[i] = (d > 0.0f) ? rsqrtf(d) : 0.0f;
  }
}

// ---------------- GEMM1: h1[N,64] = x[N,128] @ W1[128,64] (WMMA f32 16x16x4) ----------------
// block = 128 threads = 4 waves; wave w computes N-tile w (4 tiles of 16 cols);
// blockIdx.x = M-tile (16 rows).

__global__ void k_gemm1(const float* __restrict__ x, const float* __restrict__ W1,
                        float* __restrict__ h1, int N) {
  const int wave = threadIdx.x >> 5;
  const int lane = threadIdx.x & 31;
  const int hi   = lane >> 4;        // 0: lanes 0-15, 1: lanes 16-31
  const int l15  = lane & 15;
  const int row0 = blockIdx.x * 16;
  const int nc0  = wave * 16;

  int mrow = row0 + l15;
  int rclamp = (mrow < N) ? mrow : (N - 1);   // clamp OOB reads on ragged tail

  v8f c = {};
  for (int k = 0; k < DIN; k += 4) {
    const int kk = k + 2 * hi;
    v2f a;
    const float* ap = x + (long long)rclamp * DIN + kk;
    a.x = ap[0];
    a.y = ap[1];
    v2f b;
    b.x = W1[kk * DHID + nc0 + l15];
    b.y = W1[(kk + 1) * DHID + nc0 + l15];
    c = __builtin_amdgcn_wmma_f32_16x16x4_f32(false, a, false, b,
                                              (short)0, c, false, false);
  }

  const int n = nc0 + l15;
  const int mbase = row0 + 8 * hi;
#pragma unroll
  for (int j = 0; j < 8; ++j) {
    int r = mbase + j;
    if (r < N) h1[(long long)r * DHID + n] = c[j];
  }
}

// ---------------- layer-1 aggregation ----------------

__global__ void k_selfloop1(const float* __restrict__ h1, const float* __restrict__ dinv,
                            float* __restrict__ agg, int total) {
  int idx = blockIdx.x * blockDim.x + threadIdx.x;
  if (idx < total) {
    float d = dinv[idx >> 6];          // DHID == 64
    agg[idx] = h1[idx] * d * d;
  }
}

// one wave per edge, 2 features per lane
__global__ void k_scatter1(const long long* __restrict__ ei, const float* __restrict__ ew,
                           const float* __restrict__ dinv, const float* __restrict__ h1,
                           float* __restrict__ agg, int E) {
  int e = blockIdx.x * 8 + (threadIdx.x >> 5);
  int lane = threadIdx.x & 31;
  if (e >= E) return;
  int src = (int)ei[e];
  int dst = (int)ei[(long long)E + e];
  float nrm = dinv[src] * ew[e] * dinv[dst];
  int f = lane * 2;
  v2f hv = *(const v2f*)(h1 + (long long)src * DHID + f);  // 8B aligned
  atomicAdd(&agg[(long long)dst * DHID + f + 0], hv.x * nrm);
  atomicAdd(&agg[(long long)dst * DHID + f + 1], hv.y * nrm);
}

__global__ void k_relu_bias1(const float* __restrict__ agg, const float* __restrict__ b1,
                             float* __restrict__ h1r, int total) {
  int idx = blockIdx.x * blockDim.x + threadIdx.x;
  if (idx < total) h1r[idx] = fmaxf(agg[idx] + b1[idx & (DHID - 1)], 0.0f);
}

// ---------------- GEMM2: h2[N,2] = h1r[N,64] @ W2[64,2] (WMMA, N padded to 16) ----------------
// block = 256 threads = 8 waves; one wave per M-tile of 16 rows.

__global__ void k_gemm2(const float* __restrict__ h1r, const float* __restrict__ W2,
                        float* __restrict__ h2, int N) {
  const int wave = threadIdx.x >> 5;
  const int lane = threadIdx.x & 31;
  const int mtile = blockIdx.x * 8 + wave;
  if (mtile * 16 >= N) return;                 // wave-uniform exit
  const int hi  = lane >> 4;
  const int l15 = lane & 15;
  const int row0 = mtile * 16;

  int mrow = row0 + l15;
  int rclamp = (mrow < N) ? mrow : (N - 1);

  v8f c = {};
  for (int k = 0; k < DHID; k += 4) {
    const int kk = k + 2 * hi;
    v2f a;
    const float* ap = h1r + (long long)rclamp * DHID + kk;
    a.x = ap[0];
    a.y = ap[1];
    // B padded: only columns 0..1 are real
    int ncl = (l15 < DOUT) ? l15 : 0;
    float msk = (l15 < DOUT) ? 1.0f : 0.0f;
    v2f b;
    b.x = W2[kk * DOUT + ncl] * msk;
    b.y = W2[(kk + 1) * DOUT + ncl] * msk;
    c = __builtin_amdgcn_wmma_f32_16x16x4_f32(false, a, false, b,
                                              (short)0, c, false, false);
  }

  if (l15 < DOUT) {
    const int mbase = row0 + 8 * hi;
#pragma unroll
    for (int j = 0; j < 8; ++j) {
      int r = mbase + j;
      if (r < N) h2[(long long)r * DOUT + l15] = c[j];
    }
  }
}

// ---------------- layer-2 aggregation directly into d_out ----------------

__global__ void k_out_init(const float* __restrict__ h2, const float* __restrict__ dinv,
                           const float* __restrict__ b2, float* __restrict__ out, int total) {
  int idx = blockIdx.x * blockDim.x + threadIdx.x;
  if (idx < total) {
    float d = dinv[idx >> 1];
    out[idx] = h2[idx] * d * d + b2[idx & 1];
  }
}

__global__ void k_scatter2(const long long* __restrict__ ei, const float* __restrict__ ew,
                           const float* __restrict__ dinv, const float* __restrict__ h2,
                           float* __restrict__ out, int E) {
  int e = blockIdx.x * blockDim.x + threadIdx.x;
  if (e < E) {
    int src = (int)ei[e];
    int dst = (int)ei[(long long)E + e];
    float nrm = dinv[src] * ew[e] * dinv[dst];
    v2f hv = *(const v2f*)(h2 + (long long)src * DOUT);
    atomicAdd(&out[(long long)dst * DOUT + 0], hv.x * nrm);
    atomicAdd(&out[(long long)dst * DOUT + 1], hv.y * nrm);
  }
}

// ---------------- launch ----------------

extern "C" void kernel_launch(void* const* d_in, const int* in_sizes, int n_in,
                              void* d_out, int out_size, void* d_ws, size_t ws_size,
                              hipStream_t stream) {
  const float*     x  = (const float*)d_in[0];
  const long long* ei = (const long long*)d_in[1];
  const float*     ew = (const float*)d_in[2];
  const float*     W1 = (const float*)d_in[3];
  const float*     b1 = (const float*)d_in[4];
  const float*     W2 = (const float*)d_in[5];
  const float*     b2 = (const float*)d_in[6];
  float* out = (float*)d_out;

  const int N = in_sizes[0] / DIN;   // 100000
  const int E = in_sizes[2];         // 3200000

  float* ws   = (float*)d_ws;
  float* dinv = ws;                               // N
  float* h1   = dinv + N;                         // N*64 (reused for relu output)
  float* agg1 = h1 + (size_t)N * DHID;            // N*64
  float* h2   = agg1 + (size_t)N * DHID;          // N*2

  const int T = 256;
  const int mtiles = (N + 15) / 16;

  k_deg_init <<<(N + T - 1) / T, T, 0, stream>>>(dinv, N);
  k_deg_edges<<<(E + T - 1) / T, T, 0, stream>>>(ei, ew, dinv, E);
  k_dinv     <<<(N + T - 1) / T, T, 0, stream>>>(dinv, N);

  k_gemm1    <<<mtiles, 128, 0, stream>>>(x, W1, h1, N);

  k_selfloop1<<<((N * DHID) + T - 1) / T, T, 0, stream>>>(h1, dinv, agg1, N * DHID);
  k_scatter1 <<<(E + 7) / 8, 256, 0, stream>>>(ei, ew, dinv, h1, agg1, E);
  k_relu_bias1<<<((N * DHID) + T - 1) / T, T, 0, stream>>>(agg1, b1, h1, N * DHID);

  k_gemm2    <<<(mtiles + 7) / 8, 256, 0, stream>>>(h1, W2, h2, N);

  k_out_init <<<((N * DOUT) + T - 1) / T, T, 0, stream>>>(h2, dinv, b2, out, N * DOUT);
  k_scatter2 <<<(E + T - 1) / T, T, 0, stream>>>(ei, ew, dinv, h2, out, E);
}